// CombinedGeometricLoss_59768764891462
// MI455X (gfx1250) — compile-verified
//
#include <hip/hip_runtime.h>
#include <cstdint>

// CombinedGeometricLoss for MI455X (gfx1250):
//   - TDM tensor_load_to_lds stages an 18x18x8 halo brick of both grids
//     (one DMA descriptor per grid per block), s_wait_tensorcnt sync
//   - each thread computes 6 voxels (16x16x6 brick per block) from LDS
//   - wave32 shuffle reduction, f64 global atomics, finalize kernel divides.

#define DIM_ 128
#define HW_  (128 * 128)
#define DHW_ (128 * 128 * 128)
#define INTERIOR_COUNT 8001504.0   /* 4 * 126^3 */
#define COS_CLAMP_F 0.9999f        /* 1 - 1e-4 */
#define TILE_D 6                   /* computed planes per block */
#define HALO_D (TILE_D + 2)        /* staged planes per block */

#ifndef __has_builtin
#define __has_builtin(x) 0
#endif

#if __has_builtin(__builtin_amdgcn_tensor_load_to_lds)
#define HAVE_TDM 1
#else
#define HAVE_TDM 0
#endif

#if __has_builtin(__builtin_amdgcn_global_load_async_to_lds_b32)
#define HAVE_ASYNC_LDS 1
#else
#define HAVE_ASYNC_LDS 0
#endif

typedef unsigned int v4u __attribute__((ext_vector_type(4)));
typedef int v8i __attribute__((ext_vector_type(8)));
typedef int v4i __attribute__((ext_vector_type(4)));

typedef __attribute__((address_space(1))) int* gint_t;   // global (AS1)
typedef __attribute__((address_space(3))) int* lint_t;   // LDS (AS3)

#if HAVE_TDM
// One TDM descriptor: load an 18 x 18 x HALO_D (X,Y,Z) f32 brick starting at
// gtile into LDS at lds_dst (densely packed [Z][Y][X]).
// rem_* = remaining tensor extent from the tile origin (OOB -> zero-fill,
// guarantees no reads past the allocation on edge tiles).
__device__ __forceinline__ void tdm_load_tile(const float* gtile, void* lds_dst,
                                              unsigned rem_w, unsigned rem_h,
                                              unsigned rem_d) {
  const unsigned lds_addr = (unsigned)(uintptr_t)lds_dst;     // low 32 = LDS offset
  const unsigned long long ga = (unsigned long long)(uintptr_t)gtile;

  // --- D# group 0 (4 dwords) ---
  // [1:0] count=1 | [63:32] lds_addr | [120:64] global_addr | [127:126] type=2
  const v4u g0 = {
      1u,
      lds_addr,
      (unsigned)(ga & 0xFFFFFFFFu),
      (unsigned)((ga >> 32) & 0x1FFFFFFu) | (2u << 30)
  };

  // --- D# group 1 (8 dwords) ---
  // [17:16] data_size=2 (4B) | [79:48] tensor_dim0 | [111:80] tensor_dim1
  // [127:112] tile_dim0=18 | [143:128] tile_dim1=18 | [159:144] tile_dim2=HALO_D
  // [207:160] tensor_dim0_stride=128 | [255:208] tensor_dim1_stride=16384
  const v8i g1 = {
      (int)(2u << 16),
      (int)((rem_w & 0xFFFFu) << 16),            // tensor_dim0[15:0] @ bits 63:48
      (int)(((rem_w >> 16) & 0xFFFFu) | ((rem_h & 0xFFFFu) << 16)),
      (int)(((rem_h >> 16) & 0xFFFFu) | (18u << 16)),
      (int)(18u | ((unsigned)HALO_D << 16)),
      (int)128,                                  // dim0_stride[31:0]
      (int)(16384u << 16),                       // dim0_stride[47:32]=0 | dim1_stride[15:0]
      0                                          // dim1_stride[47:16]
  };

  // --- D# group 2 (4 dwords) ---
  // [31:0] tensor_dim2 | [63:32] tensor_dim3(=1 unused-safe)
  // [111:64] tensor_dim2_stride=0 | [127:112] tile_dim3=0
  const v4i g2 = { (int)rem_d, 1, 0, 0 };

  // --- D# group 3 (4 dwords) ---
  // [47:0] tensor_dim3_stride=0 | [79:48] tensor_dim4=1 | [95:80] tile_dim4=0
  const v4i g3 = { 0, (int)(1u << 16), 0, 0 };

  // VADDR4 operand: unused per ISA (set to NULL) -> zeros.
  const v8i g4 = { 0, 0, 0, 0, 0, 0, 0, 0 };

  __builtin_amdgcn_tensor_load_to_lds(g0, g1, g2, g3, g4, /*cpol=*/0);
}

__device__ __forceinline__ void wait_tensor0() {
#if __has_builtin(__builtin_amdgcn_s_wait_tensorcnt)
  __builtin_amdgcn_s_wait_tensorcnt(0);
#else
  asm volatile("s_wait_tensorcnt 0" ::: "memory");
#endif
}
#endif  // HAVE_TDM

__device__ __forceinline__ void cp_async_f32(float* lds_dst, const float* gsrc) {
#if HAVE_ASYNC_LDS
  __builtin_amdgcn_global_load_async_to_lds_b32(
      (gint_t)(uintptr_t)gsrc,
      (lint_t)(unsigned)(uintptr_t)lds_dst,
      /*offset=*/0, /*cpol=*/0);
#else
  *lds_dst = *gsrc;
#endif
}

__device__ __forceinline__ void wait_async_lds() {
#if HAVE_ASYNC_LDS
#if __has_builtin(__builtin_amdgcn_s_wait_asynccnt)
  __builtin_amdgcn_s_wait_asynccnt(0);
#else
  asm volatile("s_wait_asynccnt 0" ::: "memory");
#endif
#endif
}

__global__ void geo_loss_init_kernel(double* __restrict__ acc) {
  if (threadIdx.x < 3) acc[threadIdx.x] = 0.0;
}

__global__ __launch_bounds__(256)
void geo_loss_main_kernel(const float* __restrict__ pred,
                          const float* __restrict__ gt,
                          double* __restrict__ acc) {
  // Halo brick: HALO_D planes x 18 x 18, densely packed (TDM layout).
  __shared__ __align__(16) float sP[HALO_D][18][18];
  __shared__ __align__(16) float sG[HALO_D][18][18];
  __shared__ float red[3][8];

  const int t    = threadIdx.x;
  const int tile = blockIdx.x;            // 8x8 tiles over the 126x126 interior
  const int h0   = 1 + ((tile >> 3) << 4);
  const int w0   = 1 + ((tile & 7) << 4);
  const int d0   = 1 + (int)blockIdx.y * TILE_D;   // first computed plane, 1..121
  const int b    = blockIdx.z;            // 0..3

  const float* Pb = pred + (size_t)b * DHW_;
  const float* Gb = gt   + (size_t)b * DHW_;

#if HAVE_TDM
  if (t < 32) {   // one wave issues both DMAs (tensor ops ignore EXEC)
    const size_t toff = (size_t)(d0 - 1) * HW_ + (size_t)(h0 - 1) * DIM_ + (w0 - 1);
    const unsigned rem_w = (unsigned)(129 - w0);   // remaining extent from tile origin
    const unsigned rem_h = (unsigned)(129 - h0);
    const unsigned rem_d = (unsigned)(129 - d0);   // >= HALO_D always
    tdm_load_tile(Pb + toff, (void*)sP, rem_w, rem_h, rem_d);
    tdm_load_tile(Gb + toff, (void*)sG, rem_w, rem_h, rem_d);
    wait_tensor0();
  }
  __syncthreads();
#else
  // Per-lane async staging fallback (HALO_D*18*18 = 2592 copies per grid).
  for (int idx = t; idx < HALO_D * 324; idx += 256) {
    const int pz = idx / 324;             // 18*18 = 324
    const int r  = idx - pz * 324;
    const int py = r / 18;
    const int px = r - py * 18;
    int hh = h0 - 1 + py; hh = hh > 127 ? 127 : hh;
    int ww = w0 - 1 + px; ww = ww > 127 ? 127 : ww;
    const size_t g = (size_t)(d0 - 1 + pz) * HW_ + (size_t)hh * DIM_ + ww;
    cp_async_f32(&sP[pz][py][px], Pb + g);
    cp_async_f32(&sG[pz][py][px], Gb + g);
  }
  wait_async_lds();
  __syncthreads();
#endif

  const int tx = t & 15, ty = t >> 4;
  const int h = h0 + ty, w = w0 + tx;
  float eik = 0.f, nrm = 0.f, cnt = 0.f;
  if (h <= 126 && w <= 126) {
    const int lx = tx + 1, ly = ty + 1;
#pragma unroll
    for (int dz = 0; dz < TILE_D; ++dz) {
      const int pc = dz + 1;              // center plane in the halo brick
      // Central diffs / (2*SPACING)
      const float pW = (sP[pc][ly][lx + 1] - sP[pc][ly][lx - 1]) * 0.5f;
      const float pD = (sP[pc + 1][ly][lx] - sP[pc - 1][ly][lx]) * 0.5f;
      const float pH = (sP[pc][ly + 1][lx] - sP[pc][ly - 1][lx]) * 0.5f;
      float pn = sqrtf(pW * pW + pD * pD + pH * pH);
      pn = fminf(fmaxf(pn, 1e-4f), 10.0f);
      const float dm = pn - 1.0f;
      eik += dm * dm;

      const float gc = sG[pc][ly][lx];
      if (fabsf(gc) < 2.0f) {             // narrow band
        const float gW = (sG[pc][ly][lx + 1] - sG[pc][ly][lx - 1]) * 0.5f;
        const float gD = (sG[pc + 1][ly][lx] - sG[pc - 1][ly][lx]) * 0.5f;
        const float gH = (sG[pc][ly + 1][lx] - sG[pc][ly - 1][lx]) * 0.5f;
        float gn = sqrtf(gW * gW + gD * gD + gH * gH);
        gn = fminf(fmaxf(gn, 1e-4f), 10.0f);
        float cs = (pW * gW + pD * gD + pH * gH) / ((pn + 1e-8f) * gn);
        cs = fminf(fmaxf(cs, -COS_CLAMP_F), COS_CLAMP_F);
        nrm += 1.0f - cs;
        cnt += 1.0f;
      }
    }
  }

  // wave32 tree reduction
#pragma unroll
  for (int off = 16; off > 0; off >>= 1) {
    eik += __shfl_down(eik, off, 32);
    nrm += __shfl_down(nrm, off, 32);
    cnt += __shfl_down(cnt, off, 32);
  }
  const int lane = t & 31, wid = t >> 5;
  if (lane == 0) { red[0][wid] = eik; red[1][wid] = nrm; red[2][wid] = cnt; }
  __syncthreads();
  if (t == 0) {
    float E = 0.f, N = 0.f, C = 0.f;
#pragma unroll
    for (int i = 0; i < 8; ++i) { E += red[0][i]; N += red[1][i]; C += red[2][i]; }
    atomicAdd(&acc[0], (double)E);
    atomicAdd(&acc[1], (double)N);
    atomicAdd(&acc[2], (double)C);
  }
}

__global__ void geo_loss_final_kernel(const double* __restrict__ acc,
                                      float* __restrict__ out) {
  if (threadIdx.x == 0) {
    out[0] = (float)(acc[0] / INTERIOR_COUNT);
    out[1] = (float)(acc[2] > 0.0 ? acc[1] / acc[2] : 0.0);
  }
}

extern "C" void kernel_launch(void* const* d_in, const int* in_sizes, int n_in,
                              void* d_out, int out_size, void* d_ws, size_t ws_size,
                              hipStream_t stream) {
  (void)in_sizes; (void)n_in; (void)out_size; (void)ws_size;
  const float* pred = (const float*)d_in[0];
  const float* gt   = (const float*)d_in[1];
  float* out        = (float*)d_out;
  double* acc       = (double*)d_ws;   // [eik_sum, normal_sum, band_count]

  geo_loss_init_kernel<<<1, 32, 0, stream>>>(acc);
  dim3 grid(64, 126 / TILE_D, 4);      // 64 hw-tiles x 21 d-chunks x 4 batches
  geo_loss_main_kernel<<<grid, 256, 0, stream>>>(pred, gt, acc);
  geo_loss_final_kernel<<<1, 32, 0, stream>>>(acc, out);
}